// TreeRNNCell_5557687681543
// MI455X (gfx1250) — compile-verified
//
#include <hip/hip_runtime.h>
#include <math.h>

// CDNA5 (gfx1250) TreeRNN cell.
//   out = tanh( (x@W_in + b_in)*mask  +  segment_sum(h[src]->dst) @ W_aggr + b_aggr )
// Linearity trick: segment_sum(h[src]) @ W_aggr == segment_sum( (h@W_aggr)[src] ).
// So: g = h@W_aggr (GEMM, ws); out = (x@W_in+b_in)*mask + b_aggr (GEMM);
//     out[dst] += g[src] (atomics); out = tanh(out).

typedef __attribute__((ext_vector_type(2))) float v2f;
typedef __attribute__((ext_vector_type(8))) float v8f;

#define HDIM 128
#define TILE_ROWS 32
#define LDS_STRIDE 132   // 128 + 4 pad: row-strided ds_load_b64 hits distinct banks

// ---------------------------------------------------------------------------
// GEMM: C[N,128] = A[N,128] * W[128,128], f32 WMMA 16x16x4, K-loop of 32.
// Block = 512 threads = 16 waves; block computes 32 rows x 128 cols.
// Wave w: rowTile = w>>3 (0..1), colTile = w&7 (0..7).
// ---------------------------------------------------------------------------
template <bool FUSE_BIAS_MASK>
__global__ __launch_bounds__(512) void gemm128_wmma_f32(
    const float* __restrict__ A,            // [N,128]
    const float* __restrict__ W,            // [128,128] row-major (k, n)
    const float* __restrict__ bias0,        // b_in   (FUSE only)
    const float* __restrict__ bias1,        // b_aggr (FUSE only)
    const unsigned char* __restrict__ mask, // [N]    (FUSE only)
    float* __restrict__ C,                  // [N,128]
    int N) {
  __shared__ float Alds[TILE_ROWS * LDS_STRIDE];

  const int blockRow = blockIdx.x * TILE_ROWS;
  const int tid = threadIdx.x;

  // Cooperative A-tile load: 32 rows x 32 float4 = 1024 float4, 2 per thread.
  for (int i = tid; i < TILE_ROWS * (HDIM / 4); i += 512) {
    const int r = i >> 5;   // row in tile
    const int c4 = i & 31;  // float4 index in row
    float4 v = make_float4(0.f, 0.f, 0.f, 0.f);
    const int grow = blockRow + r;
    if (grow < N)
      v = reinterpret_cast<const float4*>(A + (size_t)grow * HDIM)[c4];
    float* dst = &Alds[r * LDS_STRIDE + c4 * 4];
    dst[0] = v.x; dst[1] = v.y; dst[2] = v.z; dst[3] = v.w;
  }
  __syncthreads();

  const int wave = tid >> 5;
  const int lane = tid & 31;
  const int rowTile = wave >> 3;  // 0..1
  const int colTile = wave & 7;   // 0..7
  const int laneHi = lane >> 4;   // 0/1 -> K sub-pair (ISA 7.12.2, 32-bit A 16x4)
  const int laneLo = lane & 15;

  v8f acc = {};

  // A frag: lane holds A[row = rowTile*16+laneLo][kk + 2*laneHi + {0,1}]
  const float* aBase = &Alds[(rowTile * 16 + laneLo) * LDS_STRIDE + 2 * laneHi];
  // B frag: lane holds W[kk + 2*laneHi + {0,1}][colTile*16 + laneLo]
  const float* wBase = W + (size_t)(2 * laneHi) * HDIM + colTile * 16 + laneLo;

#pragma unroll
  for (int kk = 0; kk < HDIM; kk += 4) {
    v2f a;
    a.x = aBase[kk];                       // ds_load_b64 (8B aligned, padded stride)
    a.y = aBase[kk + 1];
    v2f b;
    b.x = wBase[(size_t)kk * HDIM];        // 64B-contiguous across lane groups; WGP$-hot
    b.y = wBase[(size_t)(kk + 1) * HDIM];
    acc = __builtin_amdgcn_wmma_f32_16x16x4_f32(
        /*neg_a=*/false, a, /*neg_b=*/false, b,
        /*c_mod=*/(short)0, acc, /*reuse_a=*/false, /*reuse_b=*/false);
  }

  // Epilogue. C/D layout: VGPR r, lanes 0-15 -> M=r, lanes 16-31 -> M=r+8.
  const int col = colTile * 16 + laneLo;
#pragma unroll
  for (int r = 0; r < 8; ++r) {
    const int row = blockRow + rowTile * 16 + laneHi * 8 + r;
    if (row < N) {
      float v = acc[r];
      if (FUSE_BIAS_MASK) {
        const float m = (float)mask[row];
        v = (v + bias0[col]) * m + bias1[col];
      }
      C[(size_t)row * HDIM + col] = v;
    }
  }
}

// ---------------------------------------------------------------------------
// Edge scatter: one wave per edge; lane reads float4 of g[src] (coalesced
// 512B row) and atomically adds into out[dst] (global_atomic_add_f32, no-rtn).
// ---------------------------------------------------------------------------
__global__ __launch_bounds__(256) void scatter_add_rows(
    const float* __restrict__ g, const int* __restrict__ src,
    const int* __restrict__ dst, float* __restrict__ out, int E) {
  const long long t = (long long)blockIdx.x * 256 + threadIdx.x;
  const long long e = t >> 5;
  if (e >= E) return;
  const int lane = (int)(t & 31);
  const int s = src[e];
  const int d = dst[e];
  const float4 v = reinterpret_cast<const float4*>(g + (size_t)s * HDIM)[lane];
  float* o = out + (size_t)d * HDIM + lane * 4;
  atomicAdd(o + 0, v.x);
  atomicAdd(o + 1, v.y);
  atomicAdd(o + 2, v.z);
  atomicAdd(o + 3, v.w);
}

// ---------------------------------------------------------------------------
// In-place tanh, float4 vectorized (v_tanh_f32 trans op on CDNA5).
// ---------------------------------------------------------------------------
__global__ __launch_bounds__(256) void tanh_inplace(float* __restrict__ out,
                                                    long long n4) {
  const long long t = (long long)blockIdx.x * 256 + threadIdx.x;
  if (t >= n4) return;
  float4* p = reinterpret_cast<float4*>(out) + t;
  float4 v = *p;
  v.x = tanhf(v.x); v.y = tanhf(v.y); v.z = tanhf(v.z); v.w = tanhf(v.w);
  *p = v;
}

extern "C" void kernel_launch(void* const* d_in, const int* in_sizes, int n_in,
                              void* d_out, int out_size, void* d_ws,
                              size_t ws_size, hipStream_t stream) {
  const float* x        = (const float*)d_in[0];
  const float* h        = (const float*)d_in[1];
  const float* W_in     = (const float*)d_in[2];
  const float* b_in     = (const float*)d_in[3];
  const float* W_aggr   = (const float*)d_in[4];
  const float* b_aggr   = (const float*)d_in[5];
  const unsigned char* mask = (const unsigned char*)d_in[6];  // jnp.bool_ = 1 byte
  const int* edge_src   = (const int*)d_in[7];
  const int* edge_dst   = (const int*)d_in[8];

  float* out = (float*)d_out;
  float* g   = (float*)d_ws;  // [N,128] f32 = h @ W_aggr  (needs N*512 bytes)

  const int N = in_sizes[6];  // mask is [N]
  const int E = in_sizes[7];  // edge_src is [E]

  const int rowBlocks = (N + TILE_ROWS - 1) / TILE_ROWS;

  // 1) g = h @ W_aggr
  gemm128_wmma_f32<false><<<rowBlocks, 512, 0, stream>>>(
      h, W_aggr, nullptr, nullptr, nullptr, g, N);

  // 2) out = (x @ W_in + b_in) * mask + b_aggr
  gemm128_wmma_f32<true><<<rowBlocks, 512, 0, stream>>>(
      x, W_in, b_in, b_aggr, mask, out, N);

  // 3) out[dst] += g[src] for every edge
  const long long scatterThreads = (long long)E * 32;
  const int scatterBlocks = (int)((scatterThreads + 255) / 256);
  scatter_add_rows<<<scatterBlocks, 256, 0, stream>>>(g, edge_src, edge_dst,
                                                      out, E);

  // 4) out = tanh(out)
  const long long n4 = (long long)N * HDIM / 4;
  const int tanhBlocks = (int)((n4 + 255) / 256);
  tanh_inplace<<<tanhBlocks, 256, 0, stream>>>(out, n4);
}